// SoftmaxStable_84679575208560
// MI455X (gfx1250) — compile-verified
//
#include <hip/hip_runtime.h>
#include <stdint.h>

// ---------------- configuration ----------------
#define NBLK  2048          // blocks for the streaming passes
#define TPB   256           // 8 wave32s per block
#define TPB2  1024          // combine kernel block size
#define NEG_INF (-__builtin_inff())

typedef float v4f __attribute__((ext_vector_type(4)));
typedef int   v4i __attribute__((ext_vector_type(4)));

// ---------------- CDNA5 async global->LDS (guarded) ----------------
#ifndef __has_builtin
#define __has_builtin(x) 0
#endif

#if __has_builtin(__builtin_amdgcn_global_load_async_to_lds_b128)
#define HAVE_ASYNC_LDS 1
#endif

#if defined(HAVE_ASYNC_LDS)
typedef __attribute__((address_space(1))) v4i gv4i_t;   // global int4
typedef __attribute__((address_space(3))) v4i lv4i_t;   // LDS int4

// Each active lane copies 16B from its global address to its LDS address.
__device__ __forceinline__ void async_copy_b128(const void* gptr, const void* lptr) {
  // integer round-trips dodge generic<->AS cast restrictions; generic LDS
  // pointers hold the LDS byte offset in the low 32 bits (ISA 10.2 aperture).
  gv4i_t* g = (gv4i_t*)(uintptr_t)gptr;
  lv4i_t* l = (lv4i_t*)(uint32_t)(uintptr_t)lptr;
  __builtin_amdgcn_global_load_async_to_lds_b128(g, l, 0, 0);
}

__device__ __forceinline__ void wait_async_le1() {
#if __has_builtin(__builtin_amdgcn_s_wait_asynccnt)
  __builtin_amdgcn_s_wait_asynccnt(1);
#else
  asm volatile("s_wait_asynccnt 0x1" ::: "memory");
#endif
  asm volatile("" ::: "memory");   // keep the LDS consume after the wait
}

__device__ __forceinline__ void wait_async_0() {
#if __has_builtin(__builtin_amdgcn_s_wait_asynccnt)
  __builtin_amdgcn_s_wait_asynccnt(0);
#else
  asm volatile("s_wait_asynccnt 0x0" ::: "memory");
#endif
  asm volatile("" ::: "memory");
}
#endif // HAVE_ASYNC_LDS

// ---------------- online (m, s) softmax algebra ----------------
// running state: m = max seen, s = sum exp(v - m) over seen elements
__device__ __forceinline__ void upd(float v, float& m, float& s) {
  if (v > m) { s *= __expf(m - v); m = v; }   // rescale-on-new-max (rare)
  s += __expf(v - m);
}

// merge two (m,s) states; safe for the (-inf, 0) identity state
__device__ __forceinline__ void merge(float& m, float& s, float m2, float s2) {
  float M = fmaxf(m, m2);
  float a = (m  == M) ? s  : s  * __expf(m  - M);
  float b = (m2 == M) ? s2 : s2 * __expf(m2 - M);
  m = M;
  s = a + b;
}

// wave32 + cross-wave (m,s) block reduction; result valid in thread 0
__device__ __forceinline__ void block_reduce_ms(float& m, float& s,
                                                float* lm, float* ls, int nwaves) {
  const int lane = threadIdx.x & 31;
  const int wave = threadIdx.x >> 5;
#pragma unroll
  for (int off = 16; off > 0; off >>= 1) {
    float m2 = __shfl_down(m, off, 32);
    float s2 = __shfl_down(s, off, 32);
    merge(m, s, m2, s2);
  }
  if (lane == 0) { lm[wave] = m; ls[wave] = s; }
  __syncthreads();
  if (wave == 0) {
    m = (lane < nwaves) ? lm[lane] : NEG_INF;
    s = (lane < nwaves) ? ls[lane] : 0.0f;
#pragma unroll
    for (int off = 16; off > 0; off >>= 1) {
      float m2 = __shfl_down(m, off, 32);
      float s2 = __shfl_down(s, off, 32);
      merge(m, s, m2, s2);
    }
  }
}

// ---------------- pass 1: per-block online (max, sum-exp) ----------------
__global__ void __launch_bounds__(TPB)
softmax_pass1(const float* __restrict__ x, long long n,
              float* __restrict__ bm, float* __restrict__ bs) {
  const int tid = threadIdx.x;
  float m = NEG_INF, s = 0.0f;

  const v4f* __restrict__ x4 = (const v4f*)x;
  const long long n4     = n >> 2;
  const long long ntiles = n4 / TPB;          // full tiles of TPB float4s
  const long long G      = gridDim.x;

#if defined(HAVE_ASYNC_LDS)
  // 2-deep async pipeline: each wave stages its own lanes' data, no barriers.
  __shared__ v4f tile[2][TPB];
  long long g = blockIdx.x;
  int buf = 0;
  if (g < ntiles)
    async_copy_b128(x4 + g * TPB + tid, &tile[0][tid]);
  for (; g < ntiles; g += G) {
    const long long gn = g + G;
    if (gn < ntiles) {
      async_copy_b128(x4 + gn * TPB + tid, &tile[buf ^ 1][tid]);
      wait_async_le1();                       // current tile done (in-order)
    } else {
      wait_async_0();
    }
    v4f v = tile[buf][tid];
    upd(v.x, m, s); upd(v.y, m, s); upd(v.z, m, s); upd(v.w, m, s);
    buf ^= 1;
  }
#else
  for (long long g = blockIdx.x; g < ntiles; g += G) {
    __builtin_prefetch(x4 + (g + 2 * G) * TPB + tid, 0, 1);
    v4f v = x4[g * TPB + tid];
    upd(v.x, m, s); upd(v.y, m, s); upd(v.z, m, s); upd(v.w, m, s);
  }
#endif

  // remainder (block 0 only; empty for n = 64Mi)
  if (blockIdx.x == 0) {
    for (long long i = ntiles * TPB + tid; i < n4; i += TPB) {
      v4f v = x4[i];
      upd(v.x, m, s); upd(v.y, m, s); upd(v.z, m, s); upd(v.w, m, s);
    }
    for (long long i = (n4 << 2) + tid; i < n; i += TPB) upd(x[i], m, s);
  }

  __shared__ float lm[TPB / 32], ls[TPB / 32];
  block_reduce_ms(m, s, lm, ls, TPB / 32);
  if (threadIdx.x == 0) { bm[blockIdx.x] = m; bs[blockIdx.x] = s; }
}

// ---------------- pass 2: combine per-block states ----------------
__global__ void __launch_bounds__(TPB2)
softmax_combine(const float* __restrict__ bm, const float* __restrict__ bs,
                float* __restrict__ gstat, int nb) {
  float m = NEG_INF, s = 0.0f;
  for (int i = threadIdx.x; i < nb; i += TPB2) merge(m, s, bm[i], bs[i]);
  __shared__ float lm[TPB2 / 32], ls[TPB2 / 32];
  block_reduce_ms(m, s, lm, ls, TPB2 / 32);
  if (threadIdx.x == 0) {
    gstat[0] = m;            // global max
    gstat[1] = 1.0f / s;     // 1 / sum(exp(x - m))
  }
}

// ---------------- pass 3: normalize ----------------
__global__ void __launch_bounds__(TPB)
softmax_pass3(const float* __restrict__ x, float* __restrict__ out,
              const float* __restrict__ gstat, long long n) {
  const float M   = gstat[0];
  const float inv = gstat[1];
  const long long gid    = (long long)blockIdx.x * TPB + threadIdx.x;
  const long long stride = (long long)gridDim.x * TPB;
  const v4f* __restrict__ x4 = (const v4f*)x;
  v4f* __restrict__ o4 = (v4f*)out;
  const long long n4 = n >> 2;

  for (long long i = gid; i < n4; i += stride) {
    __builtin_prefetch(x4 + i + 2 * stride, 0, 1);   // global_prefetch_b8
    v4f v = x4[i];
    v4f r;
    r.x = __expf(v.x - M) * inv;
    r.y = __expf(v.y - M) * inv;
    r.z = __expf(v.z - M) * inv;
    r.w = __expf(v.w - M) * inv;
    __builtin_nontemporal_store(r, o4 + i);          // keep x resident in L2
  }
  for (long long i = (n4 << 2) + gid; i < n; i += stride)
    out[i] = __expf(x[i] - M) * inv;
}

// ---------------- host entry ----------------
extern "C" void kernel_launch(void* const* d_in, const int* in_sizes, int n_in,
                              void* d_out, int out_size, void* d_ws, size_t ws_size,
                              hipStream_t stream) {
  const float* x = (const float*)d_in[0];
  float* out = (float*)d_out;
  const long long n = (long long)in_sizes[0];

  // workspace layout: [NBLK bm][NBLK bs][2 gstat]  -> 16,392 bytes
  float* bm    = (float*)d_ws;
  float* bs    = bm + NBLK;
  float* gstat = bs + NBLK;

  softmax_pass1  <<<NBLK, TPB,  0, stream>>>(x, n, bm, bs);
  softmax_combine<<<1,    TPB2, 0, stream>>>(bm, bs, gstat, NBLK);
  softmax_pass3  <<<NBLK, TPB,  0, stream>>>(x, out, gstat, n);
}